// S2LDLoss_36945308680205
// MI455X (gfx1250) — compile-verified
//
#include <hip/hip_runtime.h>

typedef float v2f __attribute__((ext_vector_type(2)));
typedef float v8f __attribute__((ext_vector_type(8)));
typedef unsigned int u32x4 __attribute__((ext_vector_type(4)));
typedef int i32x4 __attribute__((ext_vector_type(4)));
typedef int i32x8 __attribute__((ext_vector_type(8)));

#define COARSE_H 60
#define COARSE_W 80
#define MAX_L1 (COARSE_H * COARSE_W)   // 4800
#define WPB 8                          // waves per block (kernel 1)
#define TPB (WPB * 32)

// ---------------------------------------------------------------------------
// Kernel 1: pairwise-distance min/argmin via V_WMMA_F32_16X16X4_F32.
//   A row m = ( x0, y0, |p0|^2, 1 ),  B col n = (-2x1, -2y1, 1, |p1|^2 )
//   => D[m][n] = |p0_m - p1_n|^2 in one WMMA per 16x16 tile.
//
// One block = one 16-row tile of pt0; its 8 waves each sweep 1/8 of the
// column tiles (scalar, readfirstlane'd chunk bounds -> uniform s_cbranch
// loop, unrollable) and the partial (min,argmin) are merged through LDS
// (ascending chunks preserve jnp.argmin first-index tie semantics).
//
// pt1[batch] is DMA'd by the Tensor Data Mover with LDS *padding*
// (pad after every 2 DWORDs by 2 DWORDs) so each 8-byte point lands in a
// 16-byte slot; the slots are then expanded IN PLACE to (-2x,-2y,1,|p|^2),
// making the B fetch a single guard-free ds_load_b64 per tile per lane.
// Min/argmin runs branch-free on squared distances; sqrt once per row at end.
// ---------------------------------------------------------------------------
__global__ __launch_bounds__(TPB)
void dist_min_wmma_kernel(const float* __restrict__ pt0,
                          const float* __restrict__ pt1,
                          float* __restrict__ l2min,
                          int*   __restrict__ amin,
                          int L0, int L1)
{
    extern __shared__ float4 arr4[];   // L1c slots * 16B (dynamic LDS)
    __shared__ float smin[WPB][16];
    __shared__ int   sidx[WPB][16];

    const int n    = blockIdx.y;
    const int tid  = (int)threadIdx.x;
    const int wv   = __builtin_amdgcn_readfirstlane(tid >> 5); // SGPR wave id
    const int lane = tid & 31;
    const int sub  = lane & 15;
    const bool hi  = lane >= 16;
    const int L1c  = (L1 < MAX_L1) ? L1 : MAX_L1;

    // ---- TDM: DMA pt1[n] (L1 points, 8B each) into 16B-strided LDS slots ----
    if (tid == 0) {
        const unsigned long long gaddr =
            (unsigned long long)(const void*)(pt1 + (size_t)n * L1 * 2);
        const unsigned lds_base =
            (unsigned)(unsigned long long)(const void*)&arr4[0];

        u32x4 g0;
        g0[0] = 1u;                                     // count=1 (valid user desc)
        g0[1] = lds_base;                               // lds_addr (bytes)
        g0[2] = (unsigned)(gaddr & 0xffffffffu);        // global_addr[31:0]
        g0[3] = ((unsigned)(gaddr >> 32) & 0x01ffffffu) // global_addr[56:32]
                | 0x80000000u;                          // type=2 ("image")

        i32x8 g1;
        g1[0] = (int)((3u << 16)      // data_size=3 (8 bytes/elem)
                      | (1u << 20)    // pad_enable
                      | (0u << 22)    // pad_interval: after every 2 DWORDs (8B)
                      | (1u << 25)); // pad_amount: 2 DWORDs (8B) -> 16B stride
        g1[1] = (int)(((unsigned)L1c & 0xffffu) << 16); // tensor_dim0[15:0] @[63:48]
        g1[2] = (int)(((unsigned)L1c >> 16) | (1u << 16)); // dim0[31:16] | tensor_dim1=1
        g1[3] = (int)(((unsigned)L1c & 0xffffu) << 16); // tile_dim0 @[127:112]
        g1[4] = 0;                                      // tile_dim1=0, tile_dim2=0
        g1[5] = L1c;                                    // tensor_dim0_stride[31:0]
        g1[6] = 0;
        g1[7] = 0;

        const i32x4 z4 = {0, 0, 0, 0};
#if __clang_major__ >= 23
        const i32x8 z8 = {0, 0, 0, 0, 0, 0, 0, 0};
        __builtin_amdgcn_tensor_load_to_lds(g0, g1, z4, z4, z8, 0);
#else
        __builtin_amdgcn_tensor_load_to_lds(g0, g1, z4, z4, 0);
#endif
    }

    // ---- A tile: 16 rows of w_pt0 for batch n (overlaps the DMA) ----
    const int base = blockIdx.x * 16;
    int r = base + sub; if (r >= L0) r = L0 - 1;
    const float ax = pt0[(size_t)(n * L0 + r) * 2 + 0];
    const float ay = pt0[(size_t)(n * L0 + r) * 2 + 1];
    v2f A;
    A[0] = hi ? (ax * ax + ay * ay) : ax;   // K=2: |p0|^2 | K=0: x0
    A[1] = hi ? 1.0f                : ay;   // K=3: 1      | K=1: y0

    __builtin_amdgcn_s_wait_tensorcnt(0);   // wave0: DMA done; others: no-op
    __syncthreads();                        // publish DMA completion

    // ---- in-place expand: slot (x,y,_,_) -> (-2x, -2y, 1, |p|^2) ----
    for (int i = tid; i < L1c; i += TPB) {
        const float2 p = *(const float2*)&arr4[i];
        float4 t;
        t.x = -2.0f * p.x;
        t.y = -2.0f * p.y;
        t.z = 1.0f;
        t.w = p.x * p.x + p.y * p.y;
        arr4[i] = t;
    }
    __syncthreads();

    // ---- this wave's column-tile chunk (scalar bounds; tail tile peeled) ----
    const int ftiles = L1c >> 4;                       // full 16-col tiles
    const int chunk  = (ftiles + WPB - 1) / WPB;
    const int tbeg   = wv * chunk;
    const int tend   = (tbeg + chunk < ftiles) ? (tbeg + chunk) : ftiles;

    float minv[8];
    int   mini[8];
#pragma unroll
    for (int v = 0; v < 8; ++v) { minv[v] = 3.4e38f; mini[v] = 0; }

    const v8f czero = {0.f, 0.f, 0.f, 0.f, 0.f, 0.f, 0.f, 0.f};
    const float2* arrh = (const float2*)arr4;
    const int hoff = hi ? 1 : 0;

#pragma unroll 4
    for (int t = tbeg; t < tend; ++t) {
        const int col = t * 16 + sub;
        const float2 b2 = arrh[col * 2 + hoff];  // zero-VALU B fetch
        v2f B; B[0] = b2.x; B[1] = b2.y;

        v8f D = __builtin_amdgcn_wmma_f32_16x16x4_f32(
            false, A, false, B, 0, czero, false, false);

#pragma unroll
        for (int v = 0; v < 8; ++v) {
            const bool p = D[v] < minv[v];
            minv[v] = p ? D[v] : minv[v];
            mini[v] = p ? col  : mini[v];
        }
    }

    // peeled partial tail tile (only if L1 % 16 != 0), handled by last wave
    if ((L1c & 15) && wv == WPB - 1) {
        const int t   = ftiles;
        const int col = t * 16 + sub;
        const int cc  = (col < L1c) ? col : (L1c - 1);
        const float2 b2 = arrh[cc * 2 + hoff];
        v2f B; B[0] = b2.x; B[1] = b2.y;
        v8f D = __builtin_amdgcn_wmma_f32_16x16x4_f32(
            false, A, false, B, 0, czero, false, false);
#pragma unroll
        for (int v = 0; v < 8; ++v) {
            const float d2 = (col < L1c) ? D[v] : 3.4e38f;
            const bool  p  = d2 < minv[v];
            minv[v] = p ? d2  : minv[v];
            mini[v] = p ? col : mini[v];
        }
    }

    // ---- reduce across the 16 lanes of each half (columns) ----
#pragma unroll
    for (int off = 1; off <= 8; off <<= 1) {
#pragma unroll
        for (int v = 0; v < 8; ++v) {
            const float od = __shfl_xor(minv[v], off, 32);
            const int   oi = __shfl_xor(mini[v], off, 32);
            const bool  p  = (od < minv[v]) | ((od == minv[v]) & (oi < mini[v]));
            minv[v] = p ? od : minv[v];
            mini[v] = p ? oi : mini[v];
        }
    }

    // lane 0 -> rows base+0..7 (M=v); lane 16 -> rows base+8..15 (M=v+8)
    if (sub == 0) {
        const int mb = hi ? 8 : 0;
#pragma unroll
        for (int v = 0; v < 8; ++v) {
            smin[wv][mb + v] = minv[v];
            sidx[wv][mb + v] = mini[v];
        }
    }
    __syncthreads();

    // ---- merge the 8 wave chunks (ascending column chunks -> idx tiebreak) --
    if (tid < 16) {
        float bv = smin[0][tid];
        int   bi = sidx[0][tid];
#pragma unroll
        for (int w = 1; w < WPB; ++w) {
            const float ov = smin[w][tid];
            const int   oi = sidx[w][tid];
            const bool  p  = (ov < bv) | ((ov == bv) & (oi < bi));
            bv = p ? ov : bv;
            bi = p ? oi : bi;
        }
        const int row = base + tid;
        if (row < L0) {
            l2min[n * L0 + row] = sqrtf(fmaxf(bv, 0.0f));
            amin [n * L0 + row] = bi;
        }
    }
}

// ---------------------------------------------------------------------------
// Kernel 2: single-block fused reduction (deterministic, graph-capture safe).
//   loc   = masked_mean(l2_min, dist_valid)
//   rep   = (sum(s*l2*dv) - loc*sum(s*dv)) / max(cnt,1)
//   score = masked_mean((bilinear_sample - s0)^2, mask) * 2
// ---------------------------------------------------------------------------
__global__ __launch_bounds__(512)
void fused_loss_kernel(const float* __restrict__ pt0,
                       const unsigned char* __restrict__ mask,
                       const float* __restrict__ s0,
                       const float* __restrict__ s1,
                       const float* __restrict__ scale0,
                       const float* __restrict__ scale1,
                       const float* __restrict__ l2min,
                       const int*   __restrict__ amin,
                       float* __restrict__ out,
                       int N, int L0, int L1)
{
    const int tid   = (int)threadIdx.x;
    const int total = N * L0;

    float a_l2 = 0.f, a_cnt = 0.f, a_sl2 = 0.f, a_s = 0.f, a_sc = 0.f, a_cm = 0.f;

    for (int e = tid; e < total; e += (int)blockDim.x) {
        const int n = e / L0;

        const float m   = mask[e] ? 1.0f : 0.0f;
        const float thr = 8.0f * scale0[n * 2 + 0];
        const float l2  = l2min[e];
        const float dv  = (l2 <= thr) ? m : 0.0f;

        const float s1a = s1[n * L1 + amin[e]];
        const float ss  = s1a + s0[e];

        a_l2  += l2 * dv;
        a_cnt += dv;
        a_sl2 += ss * l2 * dv;
        a_s   += ss * dv;

        // ---- bilinear grid sample of score1 (60x80), zero OOB ----
        const float x  = pt0[(size_t)e * 2 + 0];
        const float y  = pt0[(size_t)e * 2 + 1];
        const float W1 = (float)COARSE_W * scale1[n * 2 + 0];
        const float H1 = (float)COARSE_H * scale1[n * 2 + 1];
        const float nx = x / ((W1 * 8.0f - 1.0f) * 0.5f) - 1.0f;
        const float ny = y / ((H1 * 8.0f - 1.0f) * 0.5f) - 1.0f;
        const float gx = (nx + 1.0f) * 0.5f * (float)(COARSE_W - 1);
        const float gy = (ny + 1.0f) * 0.5f * (float)(COARSE_H - 1);

        const float x0 = floorf(gx), y0 = floorf(gy);
        const float x1 = x0 + 1.0f,  y1 = y0 + 1.0f;
        const float wx1 = gx - x0, wx0 = 1.0f - wx1;
        const float wy1 = gy - y0, wy0 = 1.0f - wy1;

        const float* img = s1 + (size_t)n * L1;   // viewed as (60,80)
        auto corner = [&](float xi, float yi) -> float {
            const bool inb = (xi >= 0.f) && (xi <= (float)(COARSE_W - 1)) &&
                             (yi >= 0.f) && (yi <= (float)(COARSE_H - 1));
            const int xc = (int)fminf(fmaxf(xi, 0.f), (float)(COARSE_W - 1));
            const int yc = (int)fminf(fmaxf(yi, 0.f), (float)(COARSE_H - 1));
            return inb ? img[yc * COARSE_W + xc] : 0.0f;
        };
        const float res = corner(x0, y0) * wx0 * wy0 + corner(x1, y0) * wx1 * wy0 +
                          corner(x0, y1) * wx0 * wy1 + corner(x1, y1) * wx1 * wy1;
        const float d = res - s0[e];
        a_sc += d * d * m;
        a_cm += m;
    }

    __shared__ float sm[6][512];
    sm[0][tid] = a_l2;  sm[1][tid] = a_cnt; sm[2][tid] = a_sl2;
    sm[3][tid] = a_s;   sm[4][tid] = a_sc;  sm[5][tid] = a_cm;
    __syncthreads();

    for (int s = 256; s > 0; s >>= 1) {
        if (tid < s) {
#pragma unroll
            for (int k = 0; k < 6; ++k) sm[k][tid] += sm[k][tid + s];
        }
        __syncthreads();
    }

    if (tid == 0) {
        const float cnt  = fmaxf(sm[1][0], 1.0f);
        const float mean = sm[0][0] / cnt;                       // loc loss
        const float rep  = (sm[2][0] - mean * sm[3][0]) / cnt;   // rep loss
        const float cm   = fmaxf(sm[5][0], 1.0f);
        out[0] = mean;
        out[1] = rep;
        out[2] = (sm[4][0] / cm) * 2.0f;                          // score loss
    }
}

extern "C" void kernel_launch(void* const* d_in, const int* in_sizes, int n_in,
                              void* d_out, int out_size, void* d_ws, size_t ws_size,
                              hipStream_t stream) {
    (void)n_in; (void)out_size; (void)ws_size;

    const float*         pt0    = (const float*)d_in[0];          // (N,L0,2)
    const float*         pt1    = (const float*)d_in[1];          // (N,L1,2)
    const unsigned char* vmask  = (const unsigned char*)d_in[2];  // (N,L0) bool
    const float*         s0     = (const float*)d_in[3];          // (N,L0)
    const float*         s1     = (const float*)d_in[4];          // (N,L1)
    /* d_in[5] = image1 (zeros, shape carrier only) */
    const float*         scale0 = (const float*)d_in[6];          // (N,2)
    const float*         scale1 = (const float*)d_in[7];          // (N,2)

    const int N  = in_sizes[6] / 2;
    const int L0 = in_sizes[3] / N;
    const int L1 = in_sizes[4] / N;

    float* l2min = (float*)d_ws;
    int*   amin  = (int*)((char*)d_ws + (size_t)N * L0 * sizeof(float));

    const int L1c = (L1 < MAX_L1) ? L1 : MAX_L1;
    const size_t shbytes = (size_t)L1c * sizeof(float4);  // padded slot array

    const int rtiles = (L0 + 15) / 16;
    dim3 grid1(rtiles, N);
    dist_min_wmma_kernel<<<grid1, TPB, shbytes, stream>>>(pt0, pt1, l2min, amin,
                                                          L0, L1);

    fused_loss_kernel<<<1, 512, 0, stream>>>(pt0, vmask, s0, s1, scale0, scale1,
                                             l2min, amin, (float*)d_out, N, L0, L1);
}